// WorkflowGNN_65420941852800
// MI455X (gfx1250) — compile-verified
//
#include <hip/hip_runtime.h>
#include <math.h>

#define NODE_F 32
#define HID 64
#define NCLS 10
#define NHID2 32
#define NSLOPE 0.2f
#define TILES_PER_BLOCK 8   // row tiles (of 16) per GEMM block; amortizes the TDM weight stage

typedef float v2f __attribute__((ext_vector_type(2)));
typedef float v8f __attribute__((ext_vector_type(8)));
typedef unsigned u32x4 __attribute__((ext_vector_type(4)));
typedef int i32x4 __attribute__((ext_vector_type(4)));
typedef int i32x8 __attribute__((ext_vector_type(8)));

// ---------- helpers ----------
__device__ __forceinline__ unsigned fflip(float f) {
  unsigned u = __float_as_uint(f);
  return (u & 0x80000000u) ? ~u : (u | 0x80000000u);
}
__device__ __forceinline__ float funflip(unsigned u) {
  return __uint_as_float((u & 0x80000000u) ? (u & 0x7fffffffu) : ~u);
}
__device__ __forceinline__ float lrelu(float x) { return x > 0.0f ? x : NSLOPE * x; }

// ---------- init per-node state ----------
__global__ void k_init(float* __restrict__ deg, unsigned* __restrict__ emax,
                       float* __restrict__ denom, float* __restrict__ colsum, int n) {
  int i = blockIdx.x * blockDim.x + threadIdx.x;
  if (i < n) {
    deg[i] = 1.0f;            // self-loop contributes 1 to in-degree
    emax[i] = 0x007FFFFFu;    // fflip(-inf)
    denom[i] = 0.0f;
  }
  if (i < HID) colsum[i] = 0.0f;
}

__global__ void k_deg(const int* __restrict__ dst, float* __restrict__ deg, int e) {
  int i = blockIdx.x * blockDim.x + threadIdx.x;
  if (i < e) atomicAdd(&deg[dst[i]], 1.0f);
}

__global__ void k_dinv(float* __restrict__ deg, int n) {
  int i = blockIdx.x * blockDim.x + threadIdx.x;
  if (i < n) deg[i] = rsqrtf(fmaxf(deg[i], 1e-12f));
}

// ---------- WMMA f32 GEMM: Out[n x M] = A[n x K] * W[K x M] ----------
// W panel (K x M, <=16KB) is DMA'd into LDS ONCE per block by the Tensor Data
// Mover, then amortized over TILES_PER_BLOCK row tiles. Each of the M/16 waves
// computes a 16x16 C tile per row tile with V_WMMA_F32_16X16X4_F32, reading B
// fragments from LDS.
template <int K, int M>
__global__ void k_gemm(const float* __restrict__ A, const float* __restrict__ W,
                       float* __restrict__ Out, int n) {
  __shared__ float lW[K * M];

  if (threadIdx.x == 0) {  // one TDM issue per block (waves with EXEC==0 branch around)
    unsigned lds_off = (unsigned)(uintptr_t)(void*)lW;       // flat addr low 32b == LDS offset
    unsigned long long ga = (unsigned long long)(uintptr_t)W;
    u32x4 g0;
    g0.x = 1u;                                               // count=1, user descriptor
    g0.y = lds_off;                                          // lds_addr
    g0.z = (unsigned)(ga & 0xFFFFFFFFu);                     // global_addr[31:0]
    g0.w = (unsigned)((ga >> 32) & 0x1FFFFFFu) | (2u << 30); // global_addr[56:32], type=2
    i32x8 g1;
    g1[0] = (2 << 16);        // data_size = 4B
    g1[1] = (int)(M << 16);   // tensor_dim0[15:0] = M (bits 63:48)
    g1[2] = (int)(K << 16);   // tensor_dim0 hi = 0; tensor_dim1[15:0] = K (bits 95:80)
    g1[3] = (int)(M << 16);   // tensor_dim1 hi = 0; tile_dim0 = M (bits 127:112)
    g1[4] = K;                // tile_dim1 = K; tile_dim2 = 0
    g1[5] = M;                // tensor_dim0_stride[31:0] = M
    g1[6] = 0;                // stride hi, tensor_dim1_stride lo
    g1[7] = 0;
    i32x4 z4 = {0, 0, 0, 0};  // <=2D tensor: groups 2/3 unused
#if defined(__clang_major__) && (__clang_major__ >= 23)
    i32x8 z8 = {0, 0, 0, 0, 0, 0, 0, 0};
    __builtin_amdgcn_tensor_load_to_lds(g0, g1, z4, z4, z8, 0);
#else
    __builtin_amdgcn_tensor_load_to_lds(g0, g1, z4, z4, 0);
#endif
  }
  if (threadIdx.x < 32) __builtin_amdgcn_s_wait_tensorcnt(0);
  __syncthreads();

  const int lane = threadIdx.x & 31;
  const int wave = threadIdx.x >> 5;   // column tile
  const int m0 = wave * 16;
  const int half = lane >> 4;          // K-pair select: 0 -> {k,k+1}, 1 -> {k+2,k+3}
  const int l15 = lane & 15;

  for (int t = 0; t < TILES_PER_BLOCK; ++t) {
    const int n0 = (blockIdx.x * TILES_PER_BLOCK + t) * 16;
    if (n0 >= n) break;
    int row = n0 + l15;
    if (row >= n) row = n - 1;         // EXEC must stay all-1s for WMMA; clamp, don't mask
    v8f acc = {};
#pragma unroll
    for (int k = 0; k < K; k += 4) {
      const int kk = k + 2 * half;
      v2f a = *(const v2f*)(A + (size_t)row * K + kk);  // A fragment (2 K-adjacent f32)
      v2f b;
      b.x = lW[(kk + 0) * M + m0 + l15];                // B fragment from LDS
      b.y = lW[(kk + 1) * M + m0 + l15];
      acc = __builtin_amdgcn_wmma_f32_16x16x4_f32(false, a, false, b, (short)0, acc,
                                                  false, false);
    }
    const int rbase = n0 + half * 8;   // C layout: VGPR r -> row rbase+r, col m0+l15
    float* outp = Out + (size_t)rbase * M + m0 + l15;
    if (n0 + 16 <= n) {                // fast path: full tile, no per-row guards
#pragma unroll
      for (int r = 0; r < 8; ++r) outp[(size_t)r * M] = acc[r];
    } else {
#pragma unroll
      for (int r = 0; r < 8; ++r)
        if (rbase + r < n) outp[(size_t)r * M] = acc[r];
    }
  }
}

// ---------- GCN scatter: out[dst] += in[src] * dinv[src]*dinv[dst]  (wave per edge) ----------
__global__ void k_gcn_scatter(const float* __restrict__ hin, const int* __restrict__ src,
                              const int* __restrict__ dst, const float* __restrict__ dinv,
                              float* __restrict__ out, int e) {
  int gid = blockIdx.x * blockDim.x + threadIdx.x;
  int edge = gid >> 5;
  int lane = gid & 31;
  if (edge >= e) return;
  int s = src[edge], d = dst[edge];
  float norm = dinv[s] * dinv[d];
  v2f hv = *(const v2f*)(hin + (size_t)s * HID + 2 * lane);  // b64 gather
  atomicAdd(&out[(size_t)d * HID + 2 * lane + 0], hv.x * norm);
  atomicAdd(&out[(size_t)d * HID + 2 * lane + 1], hv.y * norm);
}

// ---------- GCN epilogue: agg = relu(agg + t*dinv^2 + b)  (self-loop folded in) ----------
__global__ void k_gcn_epi(float* __restrict__ agg, const float* __restrict__ t,
                          const float* __restrict__ dinv, const float* __restrict__ bias,
                          int n) {
  int i = blockIdx.x * blockDim.x + threadIdx.x;
  if (i >= n * HID) return;
  int node = i >> 6, f = i & 63;
  float di = dinv[node];
  float v = agg[i] + t[i] * di * di + bias[f];
  agg[i] = v > 0.0f ? v : 0.0f;
}

// ---------- GAT attention scalars ----------
__global__ void k_alpha(const float* __restrict__ h, const float* __restrict__ a_s,
                        const float* __restrict__ a_d, float* __restrict__ as,
                        float* __restrict__ ad, int n) {
  int i = blockIdx.x * blockDim.x + threadIdx.x;
  if (i >= n) return;
  float s = 0.0f, d = 0.0f;
#pragma unroll
  for (int k = 0; k < HID; ++k) {
    float v = h[(size_t)i * HID + k];
    s += v * a_s[k];
    d += v * a_d[k];
  }
  as[i] = s;
  ad[i] = d;
}

// ---------- GAT segment max over incoming edges (+ self-loops) ----------
__global__ void k_gat_max(const int* __restrict__ src, const int* __restrict__ dst,
                          const float* __restrict__ as, const float* __restrict__ ad,
                          unsigned* __restrict__ emax, int e, int n) {
  int i = blockIdx.x * blockDim.x + threadIdx.x;
  if (i < e) {
    int s = src[i], d = dst[i];
    atomicMax(&emax[d], fflip(lrelu(as[s] + ad[d])));
  } else if (i < e + n) {
    int v = i - e;
    atomicMax(&emax[v], fflip(lrelu(as[v] + ad[v])));
  }
}

// ---------- GAT softmax denominator ----------
__global__ void k_gat_denom(const int* __restrict__ src, const int* __restrict__ dst,
                            const float* __restrict__ as, const float* __restrict__ ad,
                            const unsigned* __restrict__ emax, float* __restrict__ denom,
                            int e, int n) {
  int i = blockIdx.x * blockDim.x + threadIdx.x;
  if (i < e) {
    int s = src[i], d = dst[i];
    atomicAdd(&denom[d], __expf(lrelu(as[s] + ad[d]) - funflip(emax[d])));
  } else if (i < e + n) {
    int v = i - e;
    atomicAdd(&denom[v], __expf(lrelu(as[v] + ad[v]) - funflip(emax[v])));
  }
}

// ---------- GAT scatter: out[dst] += in[src] * alpha  (wave per edge) ----------
__global__ void k_gat_scatter(const float* __restrict__ hin, const int* __restrict__ src,
                              const int* __restrict__ dst, const float* __restrict__ as,
                              const float* __restrict__ ad, const unsigned* __restrict__ emax,
                              const float* __restrict__ denom, float* __restrict__ out, int e) {
  int gid = blockIdx.x * blockDim.x + threadIdx.x;
  int edge = gid >> 5;
  int lane = gid & 31;
  if (edge >= e) return;
  int s = src[edge], d = dst[edge];
  float alpha = __expf(lrelu(as[s] + ad[d]) - funflip(emax[d])) / denom[d];
  v2f hv = *(const v2f*)(hin + (size_t)s * HID + 2 * lane);
  atomicAdd(&out[(size_t)d * HID + 2 * lane + 0], hv.x * alpha);
  atomicAdd(&out[(size_t)d * HID + 2 * lane + 1], hv.y * alpha);
}

// ---------- GAT epilogue: agg = relu(agg + t*alpha_self + b) ----------
__global__ void k_gat_epi(float* __restrict__ agg, const float* __restrict__ t,
                          const float* __restrict__ as, const float* __restrict__ ad,
                          const unsigned* __restrict__ emax, const float* __restrict__ denom,
                          const float* __restrict__ bias, int n) {
  int i = blockIdx.x * blockDim.x + threadIdx.x;
  if (i >= n * HID) return;
  int node = i >> 6, f = i & 63;
  float alpha = __expf(lrelu(as[node] + ad[node]) - funflip(emax[node])) / denom[node];
  float v = agg[i] + t[i] * alpha + bias[f];
  agg[i] = v > 0.0f ? v : 0.0f;
}

// ---------- pad Wopt [64x10] -> [64x16] for the WMMA head ----------
__global__ void k_padw(const float* __restrict__ Wopt, float* __restrict__ Wp) {
  int i = blockIdx.x * blockDim.x + threadIdx.x;
  if (i >= HID * 16) return;
  int k = i >> 4, m = i & 15;
  Wp[i] = (m < NCLS) ? Wopt[k * NCLS + m] : 0.0f;
}

// ---------- head epilogues ----------
__global__ void k_opt_epi(const float* __restrict__ optbuf, const float* __restrict__ bopt,
                          float* __restrict__ out_opt, int n) {
  int i = blockIdx.x * blockDim.x + threadIdx.x;
  if (i >= n * NCLS) return;
  int node = i / NCLS, c = i % NCLS;
  out_opt[i] = optbuf[(size_t)node * 16 + c] + bopt[c];
}

__global__ void k_bot_epi(const float* __restrict__ hidbuf, const float* __restrict__ bb1,
                          const float* __restrict__ Wb2, const float* __restrict__ bb2,
                          float* __restrict__ out_bot, int n) {
  int i = blockIdx.x * blockDim.x + threadIdx.x;
  if (i >= n) return;
  float z = bb2[0];
#pragma unroll
  for (int j = 0; j < NHID2; ++j) {
    float a = hidbuf[(size_t)i * NHID2 + j] + bb1[j];
    a = a > 0.0f ? a : 0.0f;
    z += a * Wb2[j];
  }
  out_bot[i] = 1.0f / (1.0f + __expf(-z));
}

// ---------- column-sum reduction for graph embedding ----------
__global__ void k_colsum(const float* __restrict__ h, float* __restrict__ colsum, int n) {
  __shared__ float s[4][HID];
  int f = threadIdx.x & 63;
  int g = threadIdx.x >> 6;  // 0..3
  int end = (blockIdx.x + 1) * 1024;
  if (end > n) end = n;
  float acc = 0.0f;
  for (int node = blockIdx.x * 1024 + g; node < end; node += 4)
    acc += h[(size_t)node * HID + f];
  s[g][f] = acc;
  __syncthreads();
  if (g == 0) atomicAdd(&colsum[f], s[0][f] + s[1][f] + s[2][f] + s[3][f]);
}

__global__ void k_emb(const float* __restrict__ colsum, float* __restrict__ emb, float invn) {
  int i = threadIdx.x;
  if (i < HID) emb[i] = colsum[i] * invn;
}

// ---------- host orchestration ----------
extern "C" void kernel_launch(void* const* d_in, const int* in_sizes, int n_in,
                              void* d_out, int out_size, void* d_ws, size_t ws_size,
                              hipStream_t stream) {
  (void)n_in; (void)out_size; (void)ws_size;
  const float* x    = (const float*)d_in[0];
  const int*   ei   = (const int*)d_in[1];
  const float* W1   = (const float*)d_in[2];
  const float* b1   = (const float*)d_in[3];
  const float* W2   = (const float*)d_in[4];
  const float* a_s  = (const float*)d_in[5];
  const float* a_d  = (const float*)d_in[6];
  const float* b2   = (const float*)d_in[7];
  const float* W3   = (const float*)d_in[8];
  const float* b3   = (const float*)d_in[9];
  const float* Wopt = (const float*)d_in[10];
  const float* bopt = (const float*)d_in[11];
  const float* Wb1  = (const float*)d_in[12];
  const float* bb1  = (const float*)d_in[13];
  const float* Wb2  = (const float*)d_in[14];
  const float* bb2  = (const float*)d_in[15];

  const int N = in_sizes[0] / NODE_F;
  const int E = in_sizes[1] / 2;
  const int* src = ei;
  const int* dst = ei + E;

  char* ws = (char*)d_ws;
  size_t off = 0;
  float* bufA     = (float*)(ws + off); off += (size_t)N * HID * sizeof(float);
  float* bufB     = (float*)(ws + off); off += (size_t)N * HID * sizeof(float);
  float* optbuf   = (float*)(ws + off); off += (size_t)N * 16 * sizeof(float);
  float* hidbuf   = (float*)(ws + off); off += (size_t)N * NHID2 * sizeof(float);
  float* dinv     = (float*)(ws + off); off += (size_t)N * sizeof(float);
  float* asrc     = (float*)(ws + off); off += (size_t)N * sizeof(float);
  float* adst     = (float*)(ws + off); off += (size_t)N * sizeof(float);
  unsigned* emaxu = (unsigned*)(ws + off); off += (size_t)N * sizeof(unsigned);
  float* denom    = (float*)(ws + off); off += (size_t)N * sizeof(float);
  float* colsum   = (float*)(ws + off); off += HID * sizeof(float);
  float* Wp       = (float*)(ws + off); off += HID * 16 * sizeof(float);

  float* out_opt = (float*)d_out;
  float* out_bot = out_opt + (size_t)N * NCLS;
  float* out_emb = out_bot + N;

  const int nb_nodes   = (N + 255) / 256;
  const int nb_nf      = (N * HID + 255) / 256;
  const int nb_edges_t = (E + 255) / 256;
  const int nb_edge_w  = (E + 7) / 8;          // wave-per-edge, 8 waves / 256-thread block
  const int nb_en      = (E + N + 255) / 256;
  const int nb_tiles   = (N + 15) / 16;
  const int nb_gemm    = (nb_tiles + TILES_PER_BLOCK - 1) / TILES_PER_BLOCK;
  const size_t hbytes  = (size_t)N * HID * sizeof(float);

  // graph statistics
  k_init<<<nb_nodes, 256, 0, stream>>>(dinv, emaxu, denom, colsum, N);
  k_deg<<<nb_edges_t, 256, 0, stream>>>(dst, dinv, E);
  k_dinv<<<nb_nodes, 256, 0, stream>>>(dinv, N);

  // layer 1: GCN(F=32 -> H=64)
  k_gemm<NODE_F, HID><<<nb_gemm, 128, 0, stream>>>(x, W1, bufA, N);
  hipMemsetAsync(bufB, 0, hbytes, stream);
  k_gcn_scatter<<<nb_edge_w, 256, 0, stream>>>(bufA, src, dst, dinv, bufB, E);
  k_gcn_epi<<<nb_nf, 256, 0, stream>>>(bufB, bufA, dinv, b1, N);

  // layer 2: GAT(H -> H)
  k_gemm<HID, HID><<<nb_gemm, 128, 0, stream>>>(bufB, W2, bufA, N);
  k_alpha<<<nb_nodes, 256, 0, stream>>>(bufA, a_s, a_d, asrc, adst, N);
  k_gat_max<<<nb_en, 256, 0, stream>>>(src, dst, asrc, adst, emaxu, E, N);
  k_gat_denom<<<nb_en, 256, 0, stream>>>(src, dst, asrc, adst, emaxu, denom, E, N);
  hipMemsetAsync(bufB, 0, hbytes, stream);
  k_gat_scatter<<<nb_edge_w, 256, 0, stream>>>(bufA, src, dst, asrc, adst, emaxu, denom,
                                               bufB, E);
  k_gat_epi<<<nb_nf, 256, 0, stream>>>(bufB, bufA, asrc, adst, emaxu, denom, b2, N);

  // layer 3: GCN(H -> H)
  k_gemm<HID, HID><<<nb_gemm, 128, 0, stream>>>(bufB, W3, bufA, N);
  hipMemsetAsync(bufB, 0, hbytes, stream);
  k_gcn_scatter<<<nb_edge_w, 256, 0, stream>>>(bufA, src, dst, dinv, bufB, E);
  k_gcn_epi<<<nb_nf, 256, 0, stream>>>(bufB, bufA, dinv, b3, N);

  // heads (WMMA) + graph embedding
  k_padw<<<(HID * 16 + 255) / 256, 256, 0, stream>>>(Wopt, Wp);
  k_gemm<HID, 16><<<nb_gemm, 32, 0, stream>>>(bufB, Wp, optbuf, N);
  k_gemm<HID, NHID2><<<nb_gemm, 64, 0, stream>>>(bufB, Wb1, hidbuf, N);
  k_opt_epi<<<(N * NCLS + 255) / 256, 256, 0, stream>>>(optbuf, bopt, out_opt, N);
  k_bot_epi<<<nb_nodes, 256, 0, stream>>>(hidbuf, bb1, Wb2, bb2, out_bot, N);
  k_colsum<<<(N + 1023) / 1024, 256, 0, stream>>>(bufB, colsum, N);
  k_emb<<<1, 64, 0, stream>>>(colsum, out_emb, 1.0f / (float)N);
}